// Deep_Mem_ActiveOnly_40089224741405
// MI455X (gfx1250) — compile-verified
//
#include <hip/hip_runtime.h>
#include <math.h>

#define GRID  65
#define CELLS 4225            // 65*65
#define PADW  4240            // 265 tiles * 16 cols
#define NTILE 265
#define TOPK  30

typedef float v2f __attribute__((ext_vector_type(2)));
typedef float v8f __attribute__((ext_vector_type(8)));

// Monotonic (order-preserving) unsigned encoding of float for max-selection.
__device__ __forceinline__ unsigned int ordf(float f) {
  unsigned int u = __float_as_uint(f);
  return (u & 0x80000000u) ? ~u : (u | 0x80000000u);
}

__global__ void k_zero(unsigned int* __restrict__ p, int n) {
  int i = blockIdx.x * blockDim.x + threadIdx.x;
  if (i < n) p[i] = 0u;
}

__global__ void k_hist(const float* __restrict__ tex, const float* __restrict__ pts,
                       unsigned int* __restrict__ hist, int n) {
  int i = blockIdx.x * blockDim.x + threadIdx.x;
  if (i >= n) return;
  if (tex[i] > 0.5f) {                        // strict, matches (tex > 0.5)
    float x = rintf(pts[2 * i + 0] + 32.0f);  // RNE == jnp.round (half-to-even)
    float y = rintf(pts[2 * i + 1] + 32.0f);
    int i0 = (int)fminf(fmaxf(x, 0.0f), 64.0f);
    int i1 = (int)fminf(fmaxf(y, 0.0f), 64.0f);
    atomicAdd(&hist[i0 * GRID + i1], 1u);
  }
}

extern __shared__ float smem[];

// One block = 16 consecutive cells (rows). Builds local[m][j] = mem[c0+m, j] + s_m*h[j]
// via V_WMMA_F32_16X16X4_F32 rank-1 tiles into LDS, then each wave runs a stable
// 30-round max-selection over its 4225-element row.
__global__ __launch_bounds__(512) void k_topk(const float* __restrict__ mem,
                                              const unsigned int* __restrict__ hist,
                                              float* __restrict__ num,
                                              float* __restrict__ cnt) {
  __shared__ float srow[16];
  float* hs   = smem;          // PADW floats: h as float, zero-padded
  float* rows = smem + PADW;   // 16 * PADW floats: local slab

  const int tid = threadIdx.x;
  const int c0  = blockIdx.x * 16;

  for (int j = tid; j < PADW; j += 512)
    hs[j] = (j < CELLS) ? (float)hist[j] : 0.0f;
  __syncthreads();
  if (tid < 16) {
    int c = c0 + tid;
    srow[tid] = (c < CELLS) ? hs[c] : 0.0f;
  }
  __syncthreads();

  bool any = false;
  for (int m = 0; m < 16; ++m) any |= (srow[m] > 0.0f);
  if (!any) return;  // uniform across block

  const int  wv     = tid >> 5;
  const int  ln     = tid & 31;
  const bool lolane = (ln < 16);
  const int  n      = lolane ? ln : (ln - 16);  // N (column within tile)
  const int  mrow   = lolane ? 0 : 8;           // lanes0-15 -> M0..7, lanes16-31 -> M8..15

  // Row offsets/validity are tile-invariant: hoist out of the inner loop.
  int  rowBase[8];
  bool rowOK[8];
#pragma unroll
  for (int r = 0; r < 8; ++r) {
    int row    = c0 + mrow + r;
    rowOK[r]   = (row < CELLS);
    rowBase[r] = (rowOK[r] ? row : (CELLS - 1)) * CELLS;  // clamped, branch-free
  }

  // A (16x4 f32): lanes0-15 hold K=0 (v0) / K=1 (v1); lanes16-31 hold K=2/K=3. Only K=0 nonzero.
  v2f a;
  a.x = lolane ? srow[ln] : 0.0f;   // A[M=ln][K=0] = s_M
  a.y = 0.0f;

  for (int t = wv; t < NTILE; t += 16) {
    const int  col   = t * 16 + n;
    const bool colOK = (col < CELLS);
    const int  colc  = colOK ? col : (CELLS - 1);  // clamp -> unconditional coalesced load
    // B (4x16 f32): row K=0 striped across lanes 0-15 in v0; rest zero.
    v2f b;
    b.x = lolane ? hs[col] : 0.0f;
    b.y = 0.0f;
    v8f cacc;
#pragma unroll
    for (int r = 0; r < 8; ++r) {
      float v = mem[rowBase[r] + colc];            // always in-bounds (clamped)
      cacc[r] = (rowOK[r] && colOK) ? v : 0.0f;    // v_cndmask, no exec branching
    }
    v8f d = __builtin_amdgcn_wmma_f32_16x16x4_f32(
        /*neg_a=*/false, a, /*neg_b=*/false, b,
        /*c_mod=*/(short)0, cacc, /*reuse_a=*/false, /*reuse_b=*/false);
#pragma unroll
    for (int r = 0; r < 8; ++r)
      rows[(mrow + r) * PADW + col] = colOK ? d[r] : -__builtin_huge_valf();
  }
  __syncthreads();

  // Wave wv selects top-30 of row wv (value desc, ties -> lowest index).
  const float s = srow[wv];
  if (s > 0.0f) {
    volatile float* rb = rows + wv * PADW;
    for (int k = 0; k < TOPK; ++k) {
      unsigned long long key = 0ull;
      for (int j = ln; j < PADW; j += 32) {
        float v = rb[j];
        unsigned long long kk =
            ((unsigned long long)ordf(v) << 32) |
            (unsigned long long)(0xFFFFFFFFu - (unsigned int)j);
        if (kk > key) key = kk;
      }
#pragma unroll
      for (int off = 16; off > 0; off >>= 1) {
        unsigned long long o = __shfl_xor(key, off, 32);
        if (o > key) key = o;
      }
      unsigned int jw = 0xFFFFFFFFu - (unsigned int)(key & 0xFFFFFFFFull);
      float val = rb[jw];
      if (ln == 0) {
        atomicAdd(&num[jw], s * val);  // multiplicity h[c] masked points, each adds val
        atomicAdd(&cnt[jw], s);
        rb[jw] = -__builtin_huge_valf();  // remove; same-wave LDS ops are in-order
      }
    }
  }
}

__global__ void k_final(const float* __restrict__ num, const float* __restrict__ cnt,
                        float* __restrict__ out) {
  int j = blockIdx.x * blockDim.x + threadIdx.x;
  if (j < CELLS) out[j] = (cnt[j] > 0.0f) ? (num[j] / fmaxf(cnt[j], 1.0f)) : 0.0f;
}

extern "C" void kernel_launch(void* const* d_in, const int* in_sizes, int n_in,
                              void* d_out, int out_size, void* d_ws, size_t ws_size,
                              hipStream_t stream) {
  const float* tex = (const float*)d_in[0];   // (N,1) f32
  const float* pts = (const float*)d_in[1];   // (N,2) f32
  const float* mem = (const float*)d_in[2];   // (65,65,65,65) f32
  float* out = (float*)d_out;                 // 4225 f32
  const int n = in_sizes[0];

  unsigned int* hist = (unsigned int*)d_ws;          // 4225 u32 (padded region)
  float* num = (float*)d_ws + 8192;                  // 4225 f32
  float* cnt = (float*)d_ws + 16384;                 // 4225 f32

  k_zero<<<(24576 + 255) / 256, 256, 0, stream>>>((unsigned int*)d_ws, 24576);
  k_hist<<<(n + 255) / 256, 256, 0, stream>>>(tex, pts, hist, n);
  const size_t shmem = (size_t)(PADW * 17) * sizeof(float);  // 288,320 B < 320 KB/WGP
  k_topk<<<NTILE, 512, shmem, stream>>>(mem, hist, num, cnt);
  k_final<<<(CELLS + 255) / 256, 256, 0, stream>>>(num, cnt, out);
}